// TechniquePredictor_72224170049927
// MI455X (gfx1250) — compile-verified
//
#include <hip/hip_runtime.h>
#include <hip/hip_bf16.h>

// ---------------------------------------------------------------------------
// Problem constants (from reference)
// ---------------------------------------------------------------------------
#define BB   16
#define TT   8192
#define HH   256
#define LL   513          // ph_length
#define OUTD 5
#define ROWS (BB * LL)    // 8208

typedef __attribute__((ext_vector_type(2))) float v2f;
typedef __attribute__((ext_vector_type(8))) float v8f;

// ---------------------------------------------------------------------------
// Kernel 1: per-batch inclusive cumsum of ph_bd -> mel2ph
// ---------------------------------------------------------------------------
__global__ __launch_bounds__(256) void tp_scan_kernel(const int* __restrict__ ph_bd,
                                                      int* __restrict__ mel2ph) {
    __shared__ int ssum[256];
    __shared__ int sexc[256];
    const int b   = blockIdx.x;
    const int tid = threadIdx.x;
    const int per = TT / 256;  // 32
    const int* src = ph_bd + (size_t)b * TT + tid * per;
    int s = 0;
    #pragma unroll
    for (int j = 0; j < per; ++j) s += src[j];
    ssum[tid] = s;
    __syncthreads();
    if (tid == 0) {
        int run = 0;
        for (int i = 0; i < 256; ++i) { sexc[i] = run; run += ssum[i]; }
    }
    __syncthreads();
    int run = sexc[tid];
    int* dst = mel2ph + (size_t)b * TT + tid * per;
    #pragma unroll
    for (int j = 0; j < per; ++j) { run += src[j]; dst[j] = run; }
}

// ---------------------------------------------------------------------------
// Kernel 2: zero fill
// ---------------------------------------------------------------------------
__global__ __launch_bounds__(256) void tp_zero_kernel(float* __restrict__ p, int n) {
    int i = blockIdx.x * 256 + threadIdx.x;
    if (i < n) p[i] = 0.0f;
}

// ---------------------------------------------------------------------------
// Kernel 3: fused sigmoid-gate + segment scatter-add
//   one wave = 16 frames; lane owns 8 channels
// ---------------------------------------------------------------------------
__global__ __launch_bounds__(256) void tp_attn_agg_kernel(
    const float* __restrict__ feat, const int* __restrict__ mel2ph,
    const float* __restrict__ w_attn, const float* __restrict__ b_attn,
    float* __restrict__ ph_agg, float* __restrict__ denom) {
    const int blk   = blockIdx.x;
    const int b     = blk >> 6;         // 64 chunks per batch
    const int chunk = blk & 63;
    const int tid   = threadIdx.x;
    const int wave  = tid >> 5;
    const int lane  = tid & 31;
    const int t0    = chunk * 128 + wave * 16;
    const int c     = lane * 8;

    const float4* wp0 = (const float4*)(w_attn + c);
    const float4* wp1 = (const float4*)(w_attn + HH + c);
    const float4 wa0a = wp0[0], wa0b = wp0[1];
    const float4 wa1a = wp1[0], wa1b = wp1[1];
    const float  ba0  = b_attn[0], ba1 = b_attn[1];

    const int* mp = mel2ph + (size_t)b * TT;
    float acc[8];
    #pragma unroll
    for (int j = 0; j < 8; ++j) acc[j] = 0.0f;
    float dsum = 0.0f;
    int curP = mp[t0];

    for (int f = 0; f < 16; ++f) {
        const int t = t0 + f;
        const float4* xp = (const float4*)(feat + ((size_t)b * TT + t) * HH + c);
        const float4 xa = xp[0], xb = xp[1];
        float d0 = xa.x * wa0a.x + xa.y * wa0a.y + xa.z * wa0a.z + xa.w * wa0a.w
                 + xb.x * wa0b.x + xb.y * wa0b.y + xb.z * wa0b.z + xb.w * wa0b.w;
        float d1 = xa.x * wa1a.x + xa.y * wa1a.y + xa.z * wa1a.z + xa.w * wa1a.w
                 + xb.x * wa1b.x + xb.y * wa1b.y + xb.z * wa1b.z + xb.w * wa1b.w;
        #pragma unroll
        for (int m = 16; m >= 1; m >>= 1) {
            d0 += __shfl_xor(d0, m, 32);
            d1 += __shfl_xor(d1, m, 32);
        }
        const float am = 0.5f * (1.0f / (1.0f + __expf(-(d0 + ba0))) +
                                 1.0f / (1.0f + __expf(-(d1 + ba1))));
        const int p = mp[t];
        if (p != curP) {
            float* dstP = ph_agg + ((size_t)b * LL + curP) * HH + c;
            #pragma unroll
            for (int j = 0; j < 8; ++j) atomicAdd(dstP + j, acc[j]);
            if (lane == 0) atomicAdd(denom + b * LL + curP, dsum);
            #pragma unroll
            for (int j = 0; j < 8; ++j) acc[j] = 0.0f;
            dsum = 0.0f;
            curP = p;
        }
        acc[0] += xa.x * am; acc[1] += xa.y * am; acc[2] += xa.z * am; acc[3] += xa.w * am;
        acc[4] += xb.x * am; acc[5] += xb.y * am; acc[6] += xb.z * am; acc[7] += xb.w * am;
        dsum += am;
    }
    {
        float* dstP = ph_agg + ((size_t)b * LL + curP) * HH + c;
        #pragma unroll
        for (int j = 0; j < 8; ++j) atomicAdd(dstP + j, acc[j]);
        if (lane == 0) atomicAdd(denom + b * LL + curP, dsum);
    }
}

// ---------------------------------------------------------------------------
// Kernel 4: in-place normalize by denom + compute nonpad flag per row
// ---------------------------------------------------------------------------
__global__ __launch_bounds__(256) void tp_normalize_kernel(
    float* __restrict__ ph_agg, const float* __restrict__ denom,
    float* __restrict__ nonpad) {
    __shared__ float sbuf[8];
    const int row = blockIdx.x;
    const int tid = threadIdx.x;
    const float d = denom[row] + 1e-5f;
    const size_t off = (size_t)row * HH + tid;
    const float v = ph_agg[off] / d;
    ph_agg[off] = v;
    float a = fabsf(v);
    #pragma unroll
    for (int m = 16; m >= 1; m >>= 1) a += __shfl_xor(a, m, 32);
    if ((tid & 31) == 0) sbuf[tid >> 5] = a;
    __syncthreads();
    if (tid == 0) {
        float s = 0.0f;
        #pragma unroll
        for (int i = 0; i < 8; ++i) s += sbuf[i];
        nonpad[row] = (s > 0.0f) ? 1.0f : 0.0f;
    }
}

// ---------------------------------------------------------------------------
// Kernel 5: per-row LayerNorm over 256 channels (optional nonpad multiply)
// ---------------------------------------------------------------------------
template <bool NP>
__global__ __launch_bounds__(256) void tp_ln_kernel(
    const float* __restrict__ x, float* __restrict__ y,
    const float* __restrict__ g, const float* __restrict__ bet,
    const float* __restrict__ nonpad) {
    __shared__ float sbuf[8];
    __shared__ float smv[2];
    const int row = blockIdx.x;
    const int tid = threadIdx.x;
    const size_t off = (size_t)row * HH + tid;
    const float v = x[off];
    float s = v;
    #pragma unroll
    for (int m = 16; m >= 1; m >>= 1) s += __shfl_xor(s, m, 32);
    if ((tid & 31) == 0) sbuf[tid >> 5] = s;
    __syncthreads();
    if (tid == 0) {
        float t = 0.0f;
        #pragma unroll
        for (int i = 0; i < 8; ++i) t += sbuf[i];
        smv[0] = t * (1.0f / HH);
    }
    __syncthreads();
    const float mean = smv[0];
    const float dd = v - mean;
    float q = dd * dd;
    #pragma unroll
    for (int m = 16; m >= 1; m >>= 1) q += __shfl_xor(q, m, 32);
    if ((tid & 31) == 0) sbuf[tid >> 5] = q;
    __syncthreads();
    if (tid == 0) {
        float t = 0.0f;
        #pragma unroll
        for (int i = 0; i < 8; ++i) t += sbuf[i];
        smv[1] = t * (1.0f / HH);
    }
    __syncthreads();
    float o = dd * rsqrtf(smv[1] + 1e-5f) * g[tid] + bet[tid];
    if (NP) o *= nonpad[row];
    y[off] = o;
}

// ---------------------------------------------------------------------------
// Kernel 6: weight repack [O,I,K] -> WMMA-B-fragment-ordered float2 pairs.
//   Layout per tap: [kb(64)][grp(2)][o(256)] -> float2{W[4kb+2g, o], W[4kb+2g+1, o]}
//   A lane with kgrp=(lane>>4) loads exactly one float2, fully coalesced.
// ---------------------------------------------------------------------------
__global__ __launch_bounds__(256) void tp_repack_kernel(
    const float* __restrict__ w, float2* __restrict__ wk, int taps) {
    const int idx = blockIdx.x * 256 + threadIdx.x;
    const int total = taps * 32768;            // float2 elements per tap: 64*2*256
    if (idx >= total) return;
    const int k    = idx >> 15;
    const int rem  = idx & 32767;
    const int pair = rem >> 8;                 // kb*2 + grp, 0..127
    const int o    = rem & 255;
    const int i0   = pair * 2;                 // = kb*4 + grp*2
    float2 v;
    v.x = w[((size_t)o * HH + i0) * taps + k];
    v.y = w[((size_t)o * HH + i0 + 1) * taps + k];
    wk[idx] = v;
}

// ---------------------------------------------------------------------------
// Kernel 7: conv-as-GEMM with V_WMMA_F32_16X16X4_F32
//   block: 8 waves; block owns a 32-row M tile (two 16-row accumulators per
//   wave sharing each B fragment), wave owns a 16-col N slice.
//   X tile staged into LDS with global_load_async_to_lds_b128 (ASYNCcnt).
//   MODE 0: *3^-0.5 then silu   (conv1)
//   MODE 1: residual + *nonpad  (conv2, k=1)
//   MODE 2: *nonpad             (post, k=3)
// ---------------------------------------------------------------------------
template <int TAPS, int MODE>
__global__ __launch_bounds__(256) void tp_conv_gemm_kernel(
    const float* __restrict__ X, const float2* __restrict__ Wp,
    const float* __restrict__ bias, const float* __restrict__ resid,
    const float* __restrict__ nonpad, float* __restrict__ Y) {
    __shared__ __align__(16) float sx[34 * 260];   // 34 rows, stride 1040B (65*16)
    const int tile = blockIdx.x;      // 0..16
    const int b    = blockIdx.y;      // 0..15
    const int tid  = threadIdx.x;
    const int wave = tid >> 5;
    const int lane = tid & 31;
    const int tileBase = tile * 32;

    // Stage 34 input rows (halo of 1 each side) into LDS.
    // In-range 16B chunks use the CDNA5 async global->LDS path; halo rows
    // outside [0, LL) are zero-filled with plain LDS stores.
    for (int cidx = tid; cidx < 34 * 64; cidx += 256) {
        const int r   = cidx >> 6;         // row 0..33
        const int ch4 = (cidx & 63) << 2;  // channel, multiple of 4 (16B)
        const int gr  = tileBase - 1 + r;
        float* ldst = &sx[r * 260 + ch4];
        if (gr >= 0 && gr < LL) {
            const unsigned int ldsOff = (unsigned int)(uintptr_t)ldst;
            const unsigned long long ga =
                (unsigned long long)(uintptr_t)(X + ((size_t)b * LL + gr) * HH + ch4);
            asm volatile("global_load_async_to_lds_b128 %0, %1, off"
                         :: "v"(ldsOff), "v"(ga) : "memory");
        } else {
            float4 z = make_float4(0.f, 0.f, 0.f, 0.f);
            *(float4*)ldst = z;
        }
    }
    asm volatile("s_wait_asynccnt 0x0" ::: "memory");
    __syncthreads();

    const int n0    = wave * 16;
    const int o     = n0 + (lane & 15);
    const int kgrp2 = lane >> 4;             // 0 or 1 (K half per ISA layout)
    const int m     = lane & 15;             // A-matrix M index per ISA layout
    const float bo  = bias[o];

    v8f acc0, acc1;
    #pragma unroll
    for (int r = 0; r < 8; ++r) { acc0[r] = bo; acc1[r] = bo; }

    const int base = (TAPS == 3) ? 0 : 1;
    #pragma unroll
    for (int k = 0; k < TAPS; ++k) {
        const float2* Wt = Wp + (size_t)k * 32768;
        const int lrow0 = (m + k + base) * 260;
        const int lrow1 = (m + 16 + k + base) * 260;
        for (int kb = 0; kb < 64; ++kb) {
            const int ki = kb * 4 + kgrp2 * 2;
            v2f a0, a1, bf;
            a0.x = sx[lrow0 + ki];
            a0.y = sx[lrow0 + ki + 1];
            a1.x = sx[lrow1 + ki];
            a1.y = sx[lrow1 + ki + 1];
            const float2 bv = Wt[(size_t)(kb * 2 + kgrp2) * 256 + o];
            bf.x = bv.x;
            bf.y = bv.y;
            acc0 = __builtin_amdgcn_wmma_f32_16x16x4_f32(
                false, a0, false, bf, (short)0, acc0, false, false);
            acc1 = __builtin_amdgcn_wmma_f32_16x16x4_f32(
                false, a1, false, bf, (short)0, acc1, false, false);
        }
    }

    // Epilogue: C/D layout -> VGPR r: lanes 0-15 are M=r, lanes 16-31 are M=r+8
    #pragma unroll
    for (int r = 0; r < 8; ++r) {
        const int mm = (lane < 16) ? r : (r + 8);
        #pragma unroll
        for (int half = 0; half < 2; ++half) {
            const int g = tileBase + half * 16 + mm;
            if (g < LL) {
                const size_t off = ((size_t)b * LL + g) * HH + o;
                float v = (half == 0) ? acc0[r] : acc1[r];
                if (MODE == 0) {
                    v *= 0.57735026919f;                 // 3^-0.5
                    v = v / (1.0f + __expf(-v));         // silu
                } else if (MODE == 1) {
                    v = (resid[off] + v) * nonpad[b * LL + g];
                } else {
                    v *= nonpad[b * LL + g];
                }
                Y[off] = v;
            }
        }
    }
}

// ---------------------------------------------------------------------------
// Kernel 8: final 256 -> 5 projection, one wave per row
// ---------------------------------------------------------------------------
__global__ __launch_bounds__(256) void tp_logits_kernel(
    const float* __restrict__ x, const float* __restrict__ w_out,
    const float* __restrict__ b_out, float* __restrict__ out) {
    const int tid  = threadIdx.x;
    const int wave = tid >> 5;
    const int lane = tid & 31;
    const int row  = blockIdx.x * 8 + wave;   // 1026 * 8 == 8208 exactly
    const float4* xp = (const float4*)(x + (size_t)row * HH + lane * 8);
    const float4 xa = xp[0], xb = xp[1];
    float res[OUTD];
    #pragma unroll
    for (int oo = 0; oo < OUTD; ++oo) {
        const float4* wp = (const float4*)(w_out + oo * HH + lane * 8);
        const float4 wa = wp[0], wb = wp[1];
        float d = xa.x * wa.x + xa.y * wa.y + xa.z * wa.z + xa.w * wa.w
                + xb.x * wb.x + xb.y * wb.y + xb.z * wb.z + xb.w * wb.w;
        #pragma unroll
        for (int mk = 16; mk >= 1; mk >>= 1) d += __shfl_xor(d, mk, 32);
        res[oo] = d;
    }
    if (lane == 0) {
        #pragma unroll
        for (int oo = 0; oo < OUTD; ++oo)
            out[(size_t)row * OUTD + oo] = res[oo] + b_out[oo];
    }
}

// ---------------------------------------------------------------------------
// Host launcher
// ---------------------------------------------------------------------------
extern "C" void kernel_launch(void* const* d_in, const int* in_sizes, int n_in,
                              void* d_out, int out_size, void* d_ws, size_t ws_size,
                              hipStream_t stream) {
    (void)in_sizes; (void)n_in; (void)out_size; (void)ws_size;

    const float* feat    = (const float*)d_in[0];
    const int*   ph_bd   = (const int*)d_in[1];
    // d_in[2] = ph_length scalar (compile-time 513, unused)
    const float* w_attn  = (const float*)d_in[3];
    const float* b_attn  = (const float*)d_in[4];
    const float* ln1_g   = (const float*)d_in[5];
    const float* ln1_b   = (const float*)d_in[6];
    const float* conv1_w = (const float*)d_in[7];
    const float* conv1_b = (const float*)d_in[8];
    const float* conv2_w = (const float*)d_in[9];
    const float* conv2_b = (const float*)d_in[10];
    const float* ln2_g   = (const float*)d_in[11];
    const float* ln2_b   = (const float*)d_in[12];
    const float* post_w  = (const float*)d_in[13];
    const float* post_b  = (const float*)d_in[14];
    const float* w_out   = (const float*)d_in[15];
    const float* b_out   = (const float*)d_in[16];
    float* out = (float*)d_out;

    // Workspace carve-up (256B aligned)
    char* wptr = (char*)d_ws;
    auto walloc = [&](size_t bytes) {
        char* p = wptr;
        wptr += (bytes + 255) & ~(size_t)255;
        return p;
    };
    int*    mel2ph = (int*)walloc((size_t)BB * TT * sizeof(int));
    float*  denom  = (float*)walloc((size_t)ROWS * sizeof(float));
    float*  nonpad = (float*)walloc((size_t)ROWS * sizeof(float));
    float*  P      = (float*)walloc((size_t)ROWS * HH * sizeof(float)); // ph_agg -> x0
    float*  t0     = (float*)walloc((size_t)ROWS * HH * sizeof(float));
    float*  t1     = (float*)walloc((size_t)ROWS * HH * sizeof(float));
    float2* W1     = (float2*)walloc((size_t)3 * 32768 * sizeof(float2));
    float2* W2     = (float2*)walloc((size_t)1 * 32768 * sizeof(float2));
    float2* W3     = (float2*)walloc((size_t)3 * 32768 * sizeof(float2));

    // 1. mel2ph = cumsum(ph_bd)
    tp_scan_kernel<<<BB, 256, 0, stream>>>(ph_bd, mel2ph);

    // 2. zero accumulators
    {
        const int nP = ROWS * HH;
        tp_zero_kernel<<<(nP + 255) / 256, 256, 0, stream>>>(P, nP);
        tp_zero_kernel<<<(ROWS + 255) / 256, 256, 0, stream>>>(denom, ROWS);
    }

    // 3. fused gate + scatter-add (reads all of feat once)
    tp_attn_agg_kernel<<<BB * (TT / 128), 256, 0, stream>>>(
        feat, mel2ph, w_attn, b_attn, P, denom);

    // 4. normalize in place + nonpad flags
    tp_normalize_kernel<<<ROWS, 256, 0, stream>>>(P, denom, nonpad);

    // 5. weight repacks (WMMA B-fragment order, float2 pairs)
    tp_repack_kernel<<<(3 * 32768 + 255) / 256, 256, 0, stream>>>(conv1_w, W1, 3);
    tp_repack_kernel<<<(1 * 32768 + 255) / 256, 256, 0, stream>>>(conv2_w, W2, 1);
    tp_repack_kernel<<<(3 * 32768 + 255) / 256, 256, 0, stream>>>(post_w, W3, 3);

    // 6. LN1: P -> t0
    tp_ln_kernel<false><<<ROWS, 256, 0, stream>>>(P, t0, ln1_g, ln1_b, nullptr);

    const dim3 cgrid(17, BB);   // 17 * 32 = 544 >= 513 rows per batch
    // 7. conv1 (k=3) + 3^-0.5 + silu: t0 -> t1
    tp_conv_gemm_kernel<3, 0><<<cgrid, 256, 0, stream>>>(t0, W1, conv1_b, nullptr, nullptr, t1);
    // 8. conv2 (k=1) + residual(P) + nonpad: t1 -> t0
    tp_conv_gemm_kernel<1, 1><<<cgrid, 256, 0, stream>>>(t1, W2, conv2_b, P, nonpad, t0);
    // 9. LN2 * nonpad: t0 -> t1
    tp_ln_kernel<true><<<ROWS, 256, 0, stream>>>(t0, t1, ln2_g, ln2_b, nonpad);
    // 10. post conv (k=3) * nonpad: t1 -> t0
    tp_conv_gemm_kernel<3, 2><<<cgrid, 256, 0, stream>>>(t1, W3, post_b, nullptr, nonpad, t0);

    // 11. final 256->5 projection
    tp_logits_kernel<<<ROWS / 8, 256, 0, stream>>>(t0, w_out, b_out, out);
}